// GCN_81561428951395
// MI455X (gfx1250) — compile-verified
//
#include <hip/hip_runtime.h>
#include <hip/hip_bf16.h>

typedef __attribute__((ext_vector_type(2))) float v2f;
typedef __attribute__((ext_vector_type(8))) float v8f;

#define NODES 50000
#define FEAT  128
#define NGRAPH 256
#define NCLS  10

static inline int cdiv(int a, int b) { return (a + b - 1) / b; }

// ---------------- degree / symmetric norm ----------------
__global__ void k_deg_init(float* __restrict__ deg, int n) {
  int i = blockIdx.x * blockDim.x + threadIdx.x;
  if (i < n) deg[i] = 1.0f;  // self-loop weight 1
}
__global__ void k_deg_accum(const int* __restrict__ col, const float* __restrict__ w,
                            float* __restrict__ deg, int e) {
  int i = blockIdx.x * blockDim.x + threadIdx.x;
  if (i < e) atomicAdd(&deg[col[i]], w[i]);
}
__global__ void k_dinv(const float* __restrict__ deg, float* __restrict__ dinv, int n) {
  int i = blockIdx.x * blockDim.x + threadIdx.x;
  if (i < n) { float d = deg[i]; dinv[i] = (d > 0.f) ? rsqrtf(d) : 0.f; }
}
__global__ void k_norm(const int* __restrict__ row, const int* __restrict__ col,
                       const float* __restrict__ w, const float* __restrict__ dinv,
                       float* __restrict__ nrm, int e) {
  int i = blockIdx.x * blockDim.x + threadIdx.x;
  if (i < e) nrm[i] = dinv[row[i]] * w[i] * dinv[col[i]];
}

// ---------------- FP32 WMMA GEMM: O[M,128] = (relu?)A[M,128] @ W[128,128] ----------------
// Wave w of 8 owns output columns [16w,16w+16). Its 32 B-fragments (entire W
// column slice) live in VGPRs for the whole kernel. Per 16-row M tile:
// 32 x (global_load_b64 A-frag + v_wmma_f32_16x16x4_f32).
template <bool RELU>
__global__ __launch_bounds__(256) void k_gemm_wmma(const float* __restrict__ A,
                                                   const float* __restrict__ W,
                                                   float* __restrict__ O, int mtiles) {
  const int lane = threadIdx.x & 31;
  const int wave = threadIdx.x >> 5;  // 0..7 -> N tile
  const int n0   = wave << 4;
  const int half = lane >> 4;         // 0: lanes 0-15 (K0,K1), 1: lanes 16-31 (K2,K3)
  const int ln   = lane & 15;

  v2f bf[32];
#pragma unroll
  for (int j = 0; j < 32; ++j) {      // B frag: VGPR0=K(4j+2h), VGPR1=K(4j+2h+1), N=n0+ln
    const int k0 = 4 * j + 2 * half;
    bf[j].x = W[(size_t)(k0 + 0) * 128 + n0 + ln];
    bf[j].y = W[(size_t)(k0 + 1) * 128 + n0 + ln];
  }

  for (int mt = blockIdx.x; mt < mtiles; mt += gridDim.x) {
    const float* ar = A + (size_t)(mt * 16 + ln) * 128 + 2 * half;  // A frag: M=ln
    v8f acc = {0.f, 0.f, 0.f, 0.f, 0.f, 0.f, 0.f, 0.f};
#pragma unroll
    for (int j = 0; j < 32; ++j) {
      v2f a;
      a.x = ar[4 * j + 0];
      a.y = ar[4 * j + 1];
      if (RELU) { a.x = fmaxf(a.x, 0.f); a.y = fmaxf(a.y, 0.f); }
      acc = __builtin_amdgcn_wmma_f32_16x16x4_f32(false, a, false, bf[j],
                                                  (short)0, acc, false, false);
    }
    // D layout: acc[v] -> (M = v + 8*half, N = n0 + ln)
    float* orow = O + (size_t)(mt * 16 + 8 * half) * 128 + n0 + ln;
#pragma unroll
    for (int v = 0; v < 8; ++v) orow[(size_t)v * 128] = acc[v];
  }
}

// ---------------- aggregation: Hout = T*dinv^2 (self loop) + bias, then edge scatter ----------------
__global__ void k_agg_init(const float* __restrict__ T, const float* __restrict__ dinv,
                           const float* __restrict__ bias, float* __restrict__ Hout, int n) {
  int gid = blockIdx.x * blockDim.x + threadIdx.x;  // n * 32 threads, 4 feats each
  if (gid >= n * 32) return;
  int i = gid >> 5, f = (gid & 31) << 2;
  float di = dinv[i];
  float s  = di * di;
  const float4 t = *(const float4*)(T + (size_t)i * 128 + f);
  float4 o;
  o.x = t.x * s + bias[f + 0];
  o.y = t.y * s + bias[f + 1];
  o.z = t.z * s + bias[f + 2];
  o.w = t.w * s + bias[f + 3];
  *(float4*)(Hout + (size_t)i * 128 + f) = o;
}
__global__ void k_agg_scatter(const float* __restrict__ T, const int* __restrict__ row,
                              const int* __restrict__ col, const float* __restrict__ nrm,
                              float* __restrict__ Hout, int e) {
  int gid = blockIdx.x * blockDim.x + threadIdx.x;  // e * 32 threads, 4 feats each
  if (gid >= e * 32) return;
  int ed = gid >> 5, f = (gid & 31) << 2;
  int r = row[ed], c = col[ed];
  float s = nrm[ed];
  const float4 t = *(const float4*)(T + (size_t)r * 128 + f);
  float* dst = Hout + (size_t)c * 128 + f;
  atomicAdd(dst + 0, t.x * s);
  atomicAdd(dst + 1, t.y * s);
  atomicAdd(dst + 2, t.z * s);
  atomicAdd(dst + 3, t.w * s);
}

// ---------------- pooling + classifier ----------------
__global__ void k_pool_zero(float* __restrict__ gsum, float* __restrict__ cnt) {
  int i = blockIdx.x * blockDim.x + threadIdx.x;
  if (i < NGRAPH * FEAT) gsum[i] = 0.f;
  if (i < NGRAPH) cnt[i] = 0.f;
}
__global__ void k_pool_accum(const float* __restrict__ H, const int* __restrict__ batch,
                             float* __restrict__ gsum, int n) {
  int gid = blockIdx.x * blockDim.x + threadIdx.x;
  if (gid >= n * 32) return;
  int i = gid >> 5, f = (gid & 31) << 2;
  int g = batch[i];
  const float4 t = *(const float4*)(H + (size_t)i * 128 + f);
  float* dst = gsum + (size_t)g * 128 + f;
  atomicAdd(dst + 0, t.x);
  atomicAdd(dst + 1, t.y);
  atomicAdd(dst + 2, t.z);
  atomicAdd(dst + 3, t.w);
}
__global__ void k_cnt(const int* __restrict__ batch, float* __restrict__ cnt, int n) {
  int i = blockIdx.x * blockDim.x + threadIdx.x;
  if (i < n) atomicAdd(&cnt[batch[i]], 1.f);
}
__global__ void k_final(const float* __restrict__ gsum, const float* __restrict__ cnt,
                        const float* __restrict__ Wl, const float* __restrict__ bl,
                        float* __restrict__ out) {
  int g = blockIdx.x, c = threadIdx.x;
  if (c >= NCLS) return;
  float s = 0.f;
#pragma unroll 4
  for (int f = 0; f < FEAT; ++f) s += gsum[(size_t)g * 128 + f] * Wl[f * NCLS + c];
  out[g * NCLS + c] = s / fmaxf(cnt[g], 1.f) + bl[c];
}

extern "C" void kernel_launch(void* const* d_in, const int* in_sizes, int n_in,
                              void* d_out, int out_size, void* d_ws, size_t ws_size,
                              hipStream_t stream) {
  const float* x    = (const float*)d_in[0];
  const int*   ei   = (const int*)d_in[1];   // [2, E]: row then col
  const float* ew   = (const float*)d_in[2];
  const int*   bat  = (const int*)d_in[3];
  const float* W0   = (const float*)d_in[4];
  const float* b0   = (const float*)d_in[5];
  const float* W1   = (const float*)d_in[6];
  const float* b1   = (const float*)d_in[7];
  const float* W2   = (const float*)d_in[8];
  const float* b2   = (const float*)d_in[9];
  const float* Wl   = (const float*)d_in[10];
  const float* bl   = (const float*)d_in[11];
  float*       out  = (float*)d_out;

  const int n = in_sizes[0] / FEAT;  // 50000
  const int e = in_sizes[2];         // 600000
  const int* row = ei;
  const int* col = ei + e;

  // workspace carve (floats); all offsets 16B aligned
  float* ws   = (float*)d_ws;
  size_t o    = 0;
  float* deg  = ws + o; o += (size_t)n;
  float* dinv = ws + o; o += (size_t)n;
  float* nrm  = ws + o; o += (size_t)e;
  float* T    = ws + o; o += (size_t)n * FEAT;  // GEMM output
  float* HA   = ws + o; o += (size_t)n * FEAT;  // layer buffers
  float* HB   = ws + o; o += (size_t)n * FEAT;
  float* gsum = ws + o; o += (size_t)NGRAPH * FEAT;
  float* cnt  = ws + o; o += (size_t)NGRAPH;
  (void)ws_size;

  const int BT = 256;
  const int mtiles = n / 16;       // 3125
  const int gemm_blocks = 640;     // persistent; amortize B-fragment preload

  // ---- gcn_norm ----
  k_deg_init<<<cdiv(n, BT), BT, 0, stream>>>(deg, n);
  k_deg_accum<<<cdiv(e, BT), BT, 0, stream>>>(col, ew, deg, e);
  k_dinv<<<cdiv(n, BT), BT, 0, stream>>>(deg, dinv, n);
  k_norm<<<cdiv(e, BT), BT, 0, stream>>>(row, col, ew, dinv, nrm, e);

  // ---- layer 0: h = relu(agg(x @ W0) + b0)   (relu deferred to next GEMM load) ----
  k_gemm_wmma<false><<<gemm_blocks, BT, 0, stream>>>(x, W0, T, mtiles);
  k_agg_init<<<cdiv(n * 32, BT), BT, 0, stream>>>(T, dinv, b0, HA, n);
  k_agg_scatter<<<cdiv(e * 32, BT), BT, 0, stream>>>(T, row, col, nrm, HA, e);

  // ---- layer 1 ----
  k_gemm_wmma<true><<<gemm_blocks, BT, 0, stream>>>(HA, W1, T, mtiles);
  k_agg_init<<<cdiv(n * 32, BT), BT, 0, stream>>>(T, dinv, b1, HB, n);
  k_agg_scatter<<<cdiv(e * 32, BT), BT, 0, stream>>>(T, row, col, nrm, HB, e);

  // ---- layer 2 (no relu on output) ----
  k_gemm_wmma<true><<<gemm_blocks, BT, 0, stream>>>(HB, W2, T, mtiles);
  k_agg_init<<<cdiv(n * 32, BT), BT, 0, stream>>>(T, dinv, b2, HA, n);
  k_agg_scatter<<<cdiv(e * 32, BT), BT, 0, stream>>>(T, row, col, nrm, HA, e);

  // ---- global mean pool + linear ----
  k_pool_zero<<<cdiv(NGRAPH * FEAT, BT), BT, 0, stream>>>(gsum, cnt);
  k_pool_accum<<<cdiv(n * 32, BT), BT, 0, stream>>>(HA, bat, gsum, n);
  k_cnt<<<cdiv(n, BT), BT, 0, stream>>>(bat, cnt, n);
  k_final<<<NGRAPH, 32, 0, stream>>>(gsum, cnt, Wl, bl, out);
  (void)out_size; (void)n_in;
}